// Decoder_74002286510446
// MI455X (gfx1250) — compile-verified
//
#include <hip/hip_runtime.h>
#include <math.h>

typedef __attribute__((ext_vector_type(2))) float v2f;
typedef __attribute__((ext_vector_type(8))) float v8f;

#define BATCH  128
#define PDIM   100
#define EDIM   2048
#define VDIM   10000
#define LDIM   21
#define TSTEPS 20
#define HDIM   512
#define XDIM   3072   // [emb(512) | gate*ctx(2048) | h(512)]

// ---------------------------------------------------------------------------
// Generic fp32 WMMA GEMM: C = epi(A(MxK)*B(KxN) + bias)
// block = 256 threads (8 waves); block tile 64x128; wave tile 32x32 (2x2 WMMA)
// Register double-buffered K-tiles (16 wide) + global_prefetch 2 tiles ahead.
// MODE: 0 = bias only, 1 = tanh(bias+), 2 = sigmoid(bias+)*mul[m][n]
// ---------------------------------------------------------------------------
template<int MODE>
__global__ __launch_bounds__(256)
void gemm_wmma(const float* __restrict__ A, const float* __restrict__ B,
               const float* __restrict__ bias, const float* __restrict__ mul,
               float* __restrict__ C,
               int M, int N, int K, int lda, int ldb, int ldc, int mulld)
{
    __shared__ float As[64 * 20];    // 64 rows x 16 k, padded to 20 (16B-aligned rows)
    __shared__ float Bs[16 * 132];   // 16 k x 128 n, padded to 132

    const int tid  = threadIdx.x;
    const int wave = tid >> 5;
    const int lane = tid & 31;
    const int half = lane >> 4;      // 0: lanes 0-15, 1: lanes 16-31
    const int l15  = lane & 15;
    const int wm   = (wave >> 2) * 32;   // wave M offset in block tile
    const int wn   = (wave & 3)  * 32;   // wave N offset in block tile

    const int m0 = blockIdx.y * 64;
    const int n0 = blockIdx.x * 128;

    v8f zero = {0.f,0.f,0.f,0.f,0.f,0.f,0.f,0.f};
    v8f acc[2][2];
    acc[0][0] = zero; acc[0][1] = zero; acc[1][0] = zero; acc[1][1] = zero;

    // per-thread staging assignments
    const int  ar   = tid >> 2;            // A row in tile
    const int  akq  = (tid & 3) * 4;       // A k quad
    const bool aval = (m0 + ar) < M;
    const int  br0  = tid >> 5,          bn0 = (tid & 31) * 4;          // B quad 0
    const int  br1  = (tid + 256) >> 5,  bn1 = ((tid + 256) & 31) * 4;  // B quad 1
    const bool bval0 = (n0 + bn0) < N;
    const bool bval1 = (n0 + bn1) < N;

    auto loadA = [&](int kb) -> float4 {
        float4 v = make_float4(0.f, 0.f, 0.f, 0.f);
        if (aval) v = *(const float4*)(A + (size_t)(m0 + ar) * lda + kb + akq);
        return v;
    };
    auto loadB = [&](int kb, int r, int nq, bool val) -> float4 {
        float4 v = make_float4(0.f, 0.f, 0.f, 0.f);
        if (val) v = *(const float4*)(B + (size_t)(kb + r) * ldb + n0 + nq);
        return v;
    };

    float4 ra  = loadA(0);
    float4 rb0 = loadB(0, br0, bn0, bval0);
    float4 rb1 = loadB(0, br1, bn1, bval1);

    for (int kb = 0; kb < K; kb += 16) {
        __syncthreads();   // previous tile fully consumed
        *(float4*)(&As[ar  * 20  + akq]) = ra;
        *(float4*)(&Bs[br0 * 132 + bn0]) = rb0;
        *(float4*)(&Bs[br1 * 132 + bn1]) = rb1;
        __syncthreads();   // tile visible

        // issue next tile's global loads before computing (latency hiding)
        const int kn = kb + 16;
        if (kn < K) {
            ra  = loadA(kn);
            rb0 = loadB(kn, br0, bn0, bval0);
            rb1 = loadB(kn, br1, bn1, bval1);
        }
        // prefetch two tiles ahead into cache (global_prefetch_b8)
        const int kp = kb + 32;
        if (kp < K) {
            if (aval)  __builtin_prefetch(A + (size_t)(m0 + ar)  * lda + kp + akq, 0, 1);
            if (bval0) __builtin_prefetch(B + (size_t)(kp + br0) * ldb + n0 + bn0, 0, 1);
        }

        #pragma unroll
        for (int ks = 0; ks < 4; ++ks) {
            const int k0 = ks * 4;
            v2f a[2], b[2];
            // A 16x4 frag: lanes 0-15 hold K=k0,k0+1; lanes 16-31 hold K=k0+2,k0+3
            #pragma unroll
            for (int tm = 0; tm < 2; ++tm) {
                int row = wm + tm * 16 + l15;
                a[tm].x = As[row * 20 + k0 + 2 * half + 0];
                a[tm].y = As[row * 20 + k0 + 2 * half + 1];
            }
            // B 4x16 frag (mirrored striping)
            #pragma unroll
            for (int tn = 0; tn < 2; ++tn) {
                int col = wn + tn * 16 + l15;
                b[tn].x = Bs[(k0 + 2 * half + 0) * 132 + col];
                b[tn].y = Bs[(k0 + 2 * half + 1) * 132 + col];
            }
            #pragma unroll
            for (int tm = 0; tm < 2; ++tm)
                #pragma unroll
                for (int tn = 0; tn < 2; ++tn)
                    acc[tm][tn] = __builtin_amdgcn_wmma_f32_16x16x4_f32(
                        false, a[tm], false, b[tn], (short)0, acc[tm][tn],
                        false, false);
        }
    }

    // epilogue: C VGPR g holds row (g + 8*half), col = l15 within 16x16 tile
    #pragma unroll
    for (int tm = 0; tm < 2; ++tm) {
        #pragma unroll
        for (int tn = 0; tn < 2; ++tn) {
            int rb = m0 + wm + tm * 16 + half * 8;
            int cb = n0 + wn + tn * 16 + l15;
            if (cb < N) {
                #pragma unroll
                for (int g = 0; g < 8; ++g) {
                    int m = rb + g;
                    if (m < M) {
                        float v = acc[tm][tn][g] + bias[cb];
                        if (MODE == 1) v = tanhf(v);
                        if (MODE == 2) v = (1.f / (1.f + expf(-v))) *
                                           mul[(size_t)m * mulld + cb];
                        C[(size_t)m * ldc + cb] = v;
                    }
                }
            }
        }
    }
}

// ---------------------------------------------------------------------------
// wave32 reductions
// ---------------------------------------------------------------------------
__device__ inline float wave_sum(float v) {
    #pragma unroll
    for (int o = 16; o > 0; o >>= 1) v += __shfl_xor(v, o, 32);
    return v;
}
__device__ inline float wave_max(float v) {
    #pragma unroll
    for (int o = 16; o > 0; o >>= 1) v = fmaxf(v, __shfl_xor(v, o, 32));
    return v;
}

// e[b,p] = sum_h tanh(enc[b,p,h]+hW[b,h])*v_att[h] + b_v; alpha = softmax_p(e)
__global__ __launch_bounds__(256)
void attn_softmax(const float* __restrict__ enc_proj, const float* __restrict__ hW,
                  const float* __restrict__ v_att, const float* __restrict__ b_v_att,
                  float* __restrict__ alpha_ws, float* __restrict__ alpha_out, int t)
{
    __shared__ float sh_hw[HDIM];
    __shared__ float sh_v[HDIM];
    __shared__ float sh_e[PDIM];
    const int b = blockIdx.x, tid = threadIdx.x;
    for (int i = tid; i < HDIM; i += 256) { sh_hw[i] = hW[b * HDIM + i]; sh_v[i] = v_att[i]; }
    __syncthreads();
    const int wave = tid >> 5, lane = tid & 31;
    for (int p = wave; p < PDIM; p += 8) {
        const float* row = enc_proj + ((size_t)b * PDIM + p) * HDIM;
        float s = 0.f;
        for (int h = lane; h < HDIM; h += 32) s += tanhf(row[h] + sh_hw[h]) * sh_v[h];
        s = wave_sum(s);
        if (lane == 0) sh_e[p] = s + b_v_att[0];
    }
    __syncthreads();
    if (wave == 0) {
        float m = -3.0e38f;
        for (int p = lane; p < PDIM; p += 32) m = fmaxf(m, sh_e[p]);
        m = wave_max(m);
        float vals[4]; float sum = 0.f; int cnt = 0;
        for (int p = lane; p < PDIM; p += 32) { float ev = expf(sh_e[p] - m); vals[cnt++] = ev; sum += ev; }
        sum = wave_sum(sum);
        float inv = 1.f / sum; cnt = 0;
        for (int p = lane; p < PDIM; p += 32) {
            float av = vals[cnt++] * inv;
            alpha_ws[b * PDIM + p] = av;
            alpha_out[((size_t)b * TSTEPS + t) * PDIM + p] = av;
        }
    }
}

// context[b,e] = sum_p alpha[b,p] * feat[b,p,e]
__global__ __launch_bounds__(256)
void context_kernel(const float* __restrict__ feat, const float* __restrict__ alpha,
                    float* __restrict__ ctx)
{
    __shared__ float sa[PDIM];
    const int b = blockIdx.y, tid = threadIdx.x;
    if (tid < PDIM) sa[tid] = alpha[b * PDIM + tid];
    __syncthreads();
    const int e = blockIdx.x * 256 + tid;
    const float* fb = feat + (size_t)b * PDIM * EDIM + e;
    float acc = 0.f;
    for (int p = 0; p < PDIM; ++p) acc += sa[p] * fb[(size_t)p * EDIM];
    ctx[(size_t)b * EDIM + e] = acc;
}

__global__ __launch_bounds__(256)
void avg_kernel(const float* __restrict__ feat, float* __restrict__ avg)
{
    const int b = blockIdx.y;
    const int e = blockIdx.x * 256 + threadIdx.x;
    const float* fb = feat + (size_t)b * PDIM * EDIM + e;
    float acc = 0.f;
    for (int p = 0; p < PDIM; ++p) acc += fb[(size_t)p * EDIM];
    avg[(size_t)b * EDIM + e] = acc * (1.f / PDIM);
}

__global__ __launch_bounds__(256)
void emb_gather(const int* __restrict__ captions, const float* __restrict__ emb,
                float* __restrict__ emb_seq)
{
    const int bt = blockIdx.x;
    const int b = bt / TSTEPS, t = bt % TSTEPS;
    const int tok = captions[b * LDIM + t];
    for (int h = threadIdx.x; h < HDIM; h += 256)
        emb_seq[(size_t)bt * HDIM + h] = emb[(size_t)tok * HDIM + h];
}

__global__ __launch_bounds__(256)
void lstm_update(const float* __restrict__ gates, float* __restrict__ c,
                 float* __restrict__ h)
{
    const int b = blockIdx.x;
    const float* g = gates + (size_t)b * 4 * HDIM;
    for (int j = threadIdx.x; j < HDIM; j += 256) {
        float ig = g[j], fg = g[HDIM + j], gg = g[2 * HDIM + j], og = g[3 * HDIM + j];
        float si = 1.f / (1.f + expf(-ig));
        float sf = 1.f / (1.f + expf(-fg));
        float so = 1.f / (1.f + expf(-og));
        float cn = sf * c[b * HDIM + j] + si * tanhf(gg);
        c[b * HDIM + j] = cn;
        h[b * HDIM + j] = so * tanhf(cn);
    }
}

// X[b] = [emb_t | (middle written by gate GEMM) | h]
__global__ __launch_bounds__(256)
void pack_x(const float* __restrict__ emb_seq, const float* __restrict__ h,
            float* __restrict__ X, int t)
{
    const int b = blockIdx.x;
    for (int i = threadIdx.x; i < HDIM; i += 256) {
        X[(size_t)b * XDIM + i] = emb_seq[((size_t)b * TSTEPS + t) * HDIM + i];
        X[(size_t)b * XDIM + 2560 + i] = h[b * HDIM + i];
    }
}

__global__ __launch_bounds__(256)
void add_bias(const float* __restrict__ a, const float* __restrict__ b,
              float* __restrict__ o, int n)
{
    int i = blockIdx.x * 256 + threadIdx.x;
    if (i < n) o[i] = a[i] + b[i];
}

// Wcat(3072x2048) = [W_ih(2560x2048); W_hh(512x2048)], float4 copies
__global__ __launch_bounds__(256)
void pack_wcat(const float* __restrict__ W_ih, const float* __restrict__ W_hh,
               float* __restrict__ Wcat)
{
    int id  = blockIdx.x * 256 + threadIdx.x;  // float4 index, total 3072*512
    int row = id >> 9;
    int c4  = id & 511;
    const float4* src = (row < 2560) ? (const float4*)(W_ih + (size_t)row * 2048)
                                     : (const float4*)(W_hh + (size_t)(row - 2560) * 2048);
    ((float4*)(Wcat + (size_t)row * 2048))[c4] = src[c4];
}

// ---------------------------------------------------------------------------
extern "C" void kernel_launch(void* const* d_in, const int* in_sizes, int n_in,
                              void* d_out, int out_size, void* d_ws, size_t ws_size,
                              hipStream_t stream)
{
    const float* feat      = (const float*)d_in[0];
    const int*   caps      = (const int*)  d_in[1];
    const float* W_init_h  = (const float*)d_in[2];
    const float* b_init_h  = (const float*)d_in[3];
    const float* W_init_c  = (const float*)d_in[4];
    const float* b_init_c  = (const float*)d_in[5];
    const float* W_fbeta   = (const float*)d_in[6];
    const float* b_fbeta   = (const float*)d_in[7];
    const float* W_out     = (const float*)d_in[8];
    const float* b_out     = (const float*)d_in[9];
    const float* emb       = (const float*)d_in[10];
    const float* W_att_enc = (const float*)d_in[11];
    const float* b_att_enc = (const float*)d_in[12];
    const float* W_att_h   = (const float*)d_in[13];
    const float* b_att_h   = (const float*)d_in[14];
    const float* v_att     = (const float*)d_in[15];
    const float* b_v_att   = (const float*)d_in[16];
    const float* W_ih      = (const float*)d_in[17];
    const float* W_hh      = (const float*)d_in[18];
    const float* b_ih      = (const float*)d_in[19];
    const float* b_hh      = (const float*)d_in[20];

    float* preds  = (float*)d_out;                                 // (B,T,V)
    float* alphas = (float*)d_out + (size_t)BATCH * TSTEPS * VDIM; // (B,T,P)

    char* w = (char*)d_ws;
    float* avg   = (float*)w; w += 1048576;    // 128x2048
    float* h     = (float*)w; w += 262144;     // 128x512
    float* c     = (float*)w; w += 262144;     // 128x512
    float* encp  = (float*)w; w += 26214400;   // 12800x512
    float* embs  = (float*)w; w += 5242880;    // 128x20x512
    float* hW    = (float*)w; w += 262144;     // 128x512
    float* alpha = (float*)w; w += 65536;      // 128x100 (padded)
    float* ctx   = (float*)w; w += 1048576;    // 128x2048
    float* X     = (float*)w; w += 1572864;    // 128x3072
    float* gates = (float*)w; w += 1048576;    // 128x2048
    float* bg    = (float*)w; w += 8192;       // 2048
    float* Wcat  = (float*)w; w += 25165824;   // 3072x2048

    // ---- one-time prologue ----
    avg_kernel<<<dim3(EDIM / 256, BATCH), 256, 0, stream>>>(feat, avg);
    gemm_wmma<1><<<dim3(4, 2), 256, 0, stream>>>(avg, W_init_h, b_init_h, nullptr,
        h, BATCH, HDIM, EDIM, EDIM, HDIM, HDIM, 0);
    gemm_wmma<1><<<dim3(4, 2), 256, 0, stream>>>(avg, W_init_c, b_init_c, nullptr,
        c, BATCH, HDIM, EDIM, EDIM, HDIM, HDIM, 0);
    gemm_wmma<0><<<dim3(4, 200), 256, 0, stream>>>(feat, W_att_enc, b_att_enc, nullptr,
        encp, BATCH * PDIM, HDIM, EDIM, EDIM, HDIM, HDIM, 0);
    emb_gather<<<BATCH * TSTEPS, 256, 0, stream>>>(caps, emb, embs);
    add_bias<<<8, 256, 0, stream>>>(b_ih, b_hh, bg, 4 * HDIM);
    pack_wcat<<<6144, 256, 0, stream>>>(W_ih, W_hh, Wcat);

    // ---- time loop ----
    for (int t = 0; t < TSTEPS; ++t) {
        gemm_wmma<0><<<dim3(4, 2), 256, 0, stream>>>(h, W_att_h, b_att_h, nullptr,
            hW, BATCH, HDIM, HDIM, HDIM, HDIM, HDIM, 0);
        attn_softmax<<<BATCH, 256, 0, stream>>>(encp, hW, v_att, b_v_att,
            alpha, alphas, t);
        context_kernel<<<dim3(EDIM / 256, BATCH), 256, 0, stream>>>(feat, alpha, ctx);
        // gate = sigmoid(h@W_fbeta+b); X[:,512:2560] = gate*ctx
        gemm_wmma<2><<<dim3(16, 2), 256, 0, stream>>>(h, W_fbeta, b_fbeta, ctx,
            X + HDIM, BATCH, EDIM, HDIM, HDIM, EDIM, XDIM, EDIM);
        pack_x<<<BATCH, 256, 0, stream>>>(embs, h, X, t);
        gemm_wmma<0><<<dim3(16, 2), 256, 0, stream>>>(X, Wcat, bg, nullptr,
            gates, BATCH, 4 * HDIM, XDIM, XDIM, 4 * HDIM, 4 * HDIM, 0);
        lstm_update<<<BATCH, 256, 0, stream>>>(gates, c, h);
        gemm_wmma<0><<<dim3((VDIM + 127) / 128, 2), 256, 0, stream>>>(h, W_out, b_out,
            nullptr, preds + (size_t)t * VDIM, BATCH, VDIM, HDIM,
            HDIM, VDIM, TSTEPS * VDIM, 0);
    }
}